// MeanAggregator_16415365005349
// MI455X (gfx1250) — compile-verified
//
#include <hip/hip_runtime.h>
#include <stdint.h>

// MeanAggregator: out[n,:] = (1/K) * sum_k features[idx[n,k], :]
// N=100000 nodes, K=6 neighbors, V=200000 rows, D=128 f32 features.
//
// Memory-bound gather (~0.2 flop/byte): optimized for the MI455X memory
// system, not the matrix core.
//  - 1 wave (32 lanes) per node; lane l owns 16B of the 512B row -> every
//    gather is one fully coalesced 512B b128 transaction per wave.
//  - The 6 row gathers use CDNA5 async global->LDS b128 copies (ASYNCcnt):
//    no VGPR destinations, tiny register footprint, deep MLP.
//  - Feature table (102.4 MB) fits in the 192 MB L2 and is reused ~3x, so
//    feature loads keep default RT caching; the 51 MB output is written with
//    non-temporal stores so it never evicts the table.

typedef float v4f __attribute__((ext_vector_type(4)));

#define KNBR   6
#define DIM    128
#define LANES  32
#define WAVES  8          // waves per block -> 8 nodes/block, 24 KB LDS

__global__ __launch_bounds__(WAVES * LANES)
void mean_agg_kernel(const float* __restrict__ feat,
                     const int*   __restrict__ nidx,
                     float*       __restrict__ out,
                     int n_nodes)
{
    __shared__ float lds[WAVES][KNBR][DIM];   // 8 * 6 * 512B = 24 KB

    const int lane = threadIdx.x & (LANES - 1);
    const int wave = threadIdx.x >> 5;
    const int node = blockIdx.x * WAVES + wave;
    if (node >= n_nodes) return;              // wave-uniform exit

    // One index load per lane (lanes 0..5), broadcast via wave32 shuffle.
    int myrow = 0;
    if (lane < KNBR) myrow = nidx[node * KNBR + lane];

    // LDS byte offset of this lane's 16B chunk in this wave's private slot.
    const unsigned lds_lane =
        (unsigned)(uintptr_t)(&lds[wave][0][0]) + (unsigned)(lane * 16);

    // Issue all K gathers as async global->LDS b128 copies (512B/wave each).
    #pragma unroll
    for (int k = 0; k < KNBR; ++k) {
        const int row = __shfl(myrow, k, LANES);
        const uint64_t src = (uint64_t)(uintptr_t)
            (feat + ((size_t)(unsigned)row << 7) + (unsigned)(lane << 2));
        const unsigned dst = lds_lane + (unsigned)(k * DIM * 4);
        asm volatile("global_load_async_to_lds_b128 %0, %1, off"
                     :: "v"(dst), "v"(src) : "memory");
    }

    // Wait for the wave's async transfers; each wave reads only its own slot,
    // so no workgroup barrier is required.
    asm volatile("s_wait_asynccnt 0x0" ::: "memory");

    v4f acc = (v4f)0.0f;
    #pragma unroll
    for (int k = 0; k < KNBR; ++k)
        acc += *(const v4f*)(&lds[wave][k][lane * 4]);
    acc *= (1.0f / (float)KNBR);

    // Non-temporal 128-bit store: stream the output past L2 so the feature
    // table stays resident.
    __builtin_nontemporal_store(acc, (v4f*)(out + (size_t)node * DIM + lane * 4));
}

extern "C" void kernel_launch(void* const* d_in, const int* in_sizes, int n_in,
                              void* d_out, int out_size, void* d_ws, size_t ws_size,
                              hipStream_t stream) {
    const float* feat = (const float*)d_in[0];
    const int*   nidx = (const int*)d_in[1];   // int32 per harness dtype mapping
    float*       out  = (float*)d_out;

    const int n_nodes = in_sizes[1] / KNBR;    // 100000
    dim3 grid((n_nodes + WAVES - 1) / WAVES);  // 12500 blocks
    dim3 block(WAVES * LANES);                 // 256 threads = 8 wave32

    hipLaunchKernelGGL(mean_agg_kernel, grid, block, 0, stream,
                       feat, nidx, out, n_nodes);
}